// LoRALinear_76063870812300
// MI455X (gfx1250) — compile-verified
//
#include <hip/hip_runtime.h>

typedef float v2f __attribute__((ext_vector_type(2)));
typedef float v8f __attribute__((ext_vector_type(8)));

#define LDST 36  // padded LDS row stride (floats): >32, multiple of 4 (16B b128 align), odd/64 bank spread

// ---- CDNA5 async global->LDS path (guarded; falls back to sync staging) ----
#if __has_builtin(__builtin_amdgcn_global_load_async_to_lds_b128)
#define ASYNC_G2L 1
typedef int v4i_ __attribute__((vector_size(16)));   // matches builtin param type
#define AS1(p) ((v4i_*)(p))
#define AS3(p) ((__attribute__((address_space(3))) v4i_*)(p))
#if __has_builtin(__builtin_amdgcn_s_wait_asynccnt)
#define WAIT_ASYNC(n) __builtin_amdgcn_s_wait_asynccnt(n)
#else
#define WAIT_ASYNC(n) asm volatile("s_wait_asynccnt %0" ::"i"(n))
#endif
#else
#define ASYNC_G2L 0
#endif

// ---------------------------------------------------------------------------
// Kernel 1: xa[t, r] = scaling[a(t)] * sum_k x[t,k] * a_cache[rank_offset[a(t),r], k]
//           (zeroed for r >= ranks[a(t)]).  Tile: 32 tokens x 64 rank, K-loop.
// ---------------------------------------------------------------------------
#define XA_BM 32
#define XA_BN 64
#define XA_BK 32

__global__ __launch_bounds__(256) void lora_xa_kernel(
    const float* __restrict__ x, const float* __restrict__ a_cache,
    const int* __restrict__ q_start_loc, const int* __restrict__ q_seqlens,
    const int* __restrict__ adapter_ids, const int* __restrict__ rank_offset,
    const int* __restrict__ ranks, const float* __restrict__ scaling,
    float* __restrict__ xa, int K, int num_seqs, int max_rank)
{
    __shared__ float sx[2][XA_BM * LDST];
    __shared__ float sa[2][XA_BN * LDST];

    const int tid   = threadIdx.x;
    const int wave  = tid >> 5;
    const int lane  = tid & 31;
    const int laneM = lane & 15;
    const int laneK = (lane >> 4) << 1;   // 0 or 2
    const int rowW  = (wave >> 2) * 16;   // 2 wave rows
    const int colW  = (wave & 3) * 16;    // 4 wave cols
    const int t0    = blockIdx.x * XA_BM;
    const int nkb   = K / XA_BK;

    for (int s = 0; s < num_seqs; ++s) {
        const int st = q_start_loc[s];
        const int ln = q_seqlens[s];
        const int lo = st - t0;
        const int hi = lo + ln;
        if (lo >= XA_BM || hi <= 0) continue;   // block-uniform branch

        const int   a  = adapter_ids[s];
        const int   rk = ranks[a];
        const float sc = scaling[a];
        const int* ro  = rank_offset + a * max_rank;

        v8f acc = {0.f, 0.f, 0.f, 0.f, 0.f, 0.f, 0.f, 0.f};

        auto compute = [&](int buf) {
            #pragma unroll
            for (int ks = 0; ks < 8; ++ks) {
                int kk = (ks << 2) + laneK;
                v2f af, bf;
                const float* pa = &sx[buf][(rowW + laneM) * LDST + kk];
                af[0] = pa[0]; af[1] = pa[1];
                const float* pb = &sa[buf][(colW + laneM) * LDST + kk];
                bf[0] = pb[0]; bf[1] = pb[1];
                acc = __builtin_amdgcn_wmma_f32_16x16x4_f32(
                    false, af, false, bf, (short)0, acc, false, false);
            }
        };

#if ASYNC_G2L
        auto copyAsync = [&](int buf, int k0) {
            {   int r = tid >> 3, c = (tid & 7) << 2;
                __builtin_amdgcn_global_load_async_to_lds_b128(
                    AS1(&x[(size_t)(t0 + r) * K + k0 + c]),
                    AS3(&sx[buf][r * LDST + c]), 0, 0); }
            #pragma unroll
            for (int i = 0; i < 2; ++i) {
                int lin = i * 256 + tid;
                int r = lin >> 3, c = (lin & 7) << 2;
                int gr = ro[r];
                __builtin_amdgcn_global_load_async_to_lds_b128(
                    AS1(&a_cache[(size_t)gr * K + k0 + c]),
                    AS3(&sa[buf][r * LDST + c]), 0, 0);
            }
        };
        copyAsync(0, 0);
        for (int kb = 0; kb < nkb; ++kb) {
            int buf = kb & 1;
            if (kb + 1 < nkb) {
                copyAsync(buf ^ 1, (kb + 1) * XA_BK);
                WAIT_ASYNC(3);   // 3 in-order copies of current tile complete
            } else {
                WAIT_ASYNC(0);
            }
            __syncthreads();
            compute(buf);
            __syncthreads();
        }
#else
        float4 xr, ar[2];
        auto loadG = [&](int k0) {
            {   int r = tid >> 3, c = (tid & 7) << 2;
                xr = *(const float4*)&x[(size_t)(t0 + r) * K + k0 + c]; }
            #pragma unroll
            for (int i = 0; i < 2; ++i) {
                int lin = i * 256 + tid;
                int r = lin >> 3, c = (lin & 7) << 2;
                int gr = ro[r];
                ar[i] = *(const float4*)&a_cache[(size_t)gr * K + k0 + c];
            }
        };
        auto storeS = [&](int buf) {
            {   int r = tid >> 3, c = (tid & 7) << 2;
                *(float4*)&sx[buf][r * LDST + c] = xr; }
            #pragma unroll
            for (int i = 0; i < 2; ++i) {
                int lin = i * 256 + tid;
                int r = lin >> 3, c = (lin & 7) << 2;
                *(float4*)&sa[buf][r * LDST + c] = ar[i];
            }
        };
        loadG(0); storeS(0); __syncthreads();
        for (int kb = 0; kb < nkb; ++kb) {
            int buf = kb & 1;
            if (kb + 1 < nkb) loadG((kb + 1) * XA_BK);
            compute(buf);
            __syncthreads();
            if (kb + 1 < nkb) { storeS(buf ^ 1); __syncthreads(); }
        }
#endif

        // store rows belonging to this sequence; rank-mask + scaling folded in
        const int   col = colW + laneM;
        const float cm  = (col < rk) ? sc : 0.0f;
        const int rbase = rowW + ((lane >> 4) << 3);
        #pragma unroll
        for (int v = 0; v < 8; ++v) {
            int row = rbase + v;
            if (row >= lo && row < hi)
                xa[(size_t)(t0 + row) * max_rank + col] = acc[v] * cm;
        }
        __syncthreads();
    }
}

// ---------------------------------------------------------------------------
// Kernel 2: out = x @ W^T  (+ fused LoRA epilogue: xa @ B_adapter, row-masked)
// Tile 128x128, BK=32, 8 waves; wave tile 64x32 = 4x2 WMMA accumulators.
// ---------------------------------------------------------------------------
#define BM 128
#define BN 128
#define BKK 32

__global__ __launch_bounds__(256) void lora_main_kernel(
    const float* __restrict__ x, const float* __restrict__ w,
    const float* __restrict__ b_cache, const float* __restrict__ xa,
    const int* __restrict__ q_start_loc, const int* __restrict__ q_seqlens,
    const int* __restrict__ adapter_ids, const int* __restrict__ rank_offset,
    float* __restrict__ out, int K, int N, int num_seqs, int max_rank)
{
    __shared__ float sx[2][BM * LDST];
    __shared__ float sw[2][BN * LDST];

    const int tid   = threadIdx.x;
    const int wave  = tid >> 5;
    const int lane  = tid & 31;
    const int laneM = lane & 15;
    const int laneK = (lane >> 4) << 1;   // 0 or 2
    const int rowW  = (wave >> 2) * 64;   // 2 waves in M
    const int colW  = (wave & 3) * 32;    // 4 waves in N
    const int n0    = blockIdx.x * BN;
    const int t0    = blockIdx.y * BM;
    const int nkb   = K / BKK;

    v8f acc[4][2];
    const v8f vzero = {0.f, 0.f, 0.f, 0.f, 0.f, 0.f, 0.f, 0.f};
    #pragma unroll
    for (int m = 0; m < 4; ++m)
        #pragma unroll
        for (int n = 0; n < 2; ++n) acc[m][n] = vzero;

    auto compute = [&](int buf) {
        #pragma unroll
        for (int ks = 0; ks < 8; ++ks) {
            int kk = (ks << 2) + laneK;
            v2f af[4], bf[2];
            #pragma unroll
            for (int m = 0; m < 4; ++m) {
                const float* p = &sx[buf][(rowW + m * 16 + laneM) * LDST + kk];
                af[m][0] = p[0]; af[m][1] = p[1];
            }
            #pragma unroll
            for (int n = 0; n < 2; ++n) {
                const float* p = &sw[buf][(colW + n * 16 + laneM) * LDST + kk];
                bf[n][0] = p[0]; bf[n][1] = p[1];
            }
            #pragma unroll
            for (int m = 0; m < 4; ++m)
                #pragma unroll
                for (int n = 0; n < 2; ++n)
                    acc[m][n] = __builtin_amdgcn_wmma_f32_16x16x4_f32(
                        false, af[m], false, bf[n], (short)0, acc[m][n], false, false);
        }
    };

#if ASYNC_G2L
    auto copyAsync = [&](int buf, int k0) {
        #pragma unroll
        for (int i = 0; i < 4; ++i) {
            int lin = i * 256 + tid;
            int r = lin >> 3, c = (lin & 7) << 2;
            __builtin_amdgcn_global_load_async_to_lds_b128(
                AS1(&x[(size_t)(t0 + r) * K + k0 + c]),
                AS3(&sx[buf][r * LDST + c]), 0, 0);
            __builtin_amdgcn_global_load_async_to_lds_b128(
                AS1(&w[(size_t)(n0 + r) * K + k0 + c]),
                AS3(&sw[buf][r * LDST + c]), 0, 0);
        }
    };
    copyAsync(0, 0);
    for (int kb = 0; kb < nkb; ++kb) {
        int buf = kb & 1;
        if (kb + 1 < nkb) {
            copyAsync(buf ^ 1, (kb + 1) * BKK);
            WAIT_ASYNC(8);   // current tile's 8 in-order async copies complete
        } else {
            WAIT_ASYNC(0);
        }
        __syncthreads();
        compute(buf);
        __syncthreads();
    }
#else
    float4 xr[4], wr[4];
    auto loadG = [&](int k0) {
        #pragma unroll
        for (int i = 0; i < 4; ++i) {
            int lin = i * 256 + tid;
            int r = lin >> 3, c = (lin & 7) << 2;
            xr[i] = *(const float4*)&x[(size_t)(t0 + r) * K + k0 + c];
            wr[i] = *(const float4*)&w[(size_t)(n0 + r) * K + k0 + c];
        }
    };
    auto storeS = [&](int buf) {
        #pragma unroll
        for (int i = 0; i < 4; ++i) {
            int lin = i * 256 + tid;
            int r = lin >> 3, c = (lin & 7) << 2;
            *(float4*)&sx[buf][r * LDST + c] = xr[i];
            *(float4*)&sw[buf][r * LDST + c] = wr[i];
        }
    };
    loadG(0); storeS(0); __syncthreads();
    for (int kb = 0; kb < nkb; ++kb) {
        int buf = kb & 1;
        if (kb + 1 < nkb) loadG((kb + 1) * BKK);
        compute(buf);
        __syncthreads();
        if (kb + 1 < nkb) { storeS(buf ^ 1); __syncthreads(); }
    }
#endif

    // ---- fused LoRA epilogue: acc += xa_tile @ B_adapter (row-masked) ----
    const int nks = max_rank >> 2;
    for (int s = 0; s < num_seqs; ++s) {
        const int st = q_start_loc[s];
        const int ln = q_seqlens[s];
        const int lo = st - t0;
        const int hi = lo + ln;
        if (lo >= BM || hi <= 0) continue;    // block-uniform
        const int a = adapter_ids[s];
        const int* ro = rank_offset + a * max_rank;
        for (int ks = 0; ks < nks; ++ks) {
            int kk = (ks << 2) + laneK;
            int r0 = ro[kk], r1 = ro[kk + 1];
            v2f af[4], bf[2];
            #pragma unroll
            for (int m = 0; m < 4; ++m) {
                int row = rowW + m * 16 + laneM;
                float msk = (row >= lo && row < hi) ? 1.0f : 0.0f;
                const float* p = &xa[(size_t)(t0 + row) * max_rank + kk];
                af[m][0] = p[0] * msk; af[m][1] = p[1] * msk;
            }
            #pragma unroll
            for (int n = 0; n < 2; ++n) {
                int col = n0 + colW + n * 16 + laneM;
                bf[n][0] = b_cache[(size_t)r0 * N + col];
                bf[n][1] = b_cache[(size_t)r1 * N + col];
            }
            #pragma unroll
            for (int m = 0; m < 4; ++m)
                #pragma unroll
                for (int n = 0; n < 2; ++n)
                    acc[m][n] = __builtin_amdgcn_wmma_f32_16x16x4_f32(
                        false, af[m], false, bf[n], (short)0, acc[m][n], false, false);
        }
    }

    // ---- store C ----
    #pragma unroll
    for (int m = 0; m < 4; ++m) {
        int rbase = t0 + rowW + m * 16 + ((lane >> 4) << 3);
        #pragma unroll
        for (int n = 0; n < 2; ++n) {
            int col = n0 + colW + n * 16 + laneM;
            #pragma unroll
            for (int v = 0; v < 8; ++v)
                out[(size_t)(rbase + v) * N + col] = acc[m][n][v];
        }
    }
}

// ---------------------------------------------------------------------------
extern "C" void kernel_launch(void* const* d_in, const int* in_sizes, int n_in,
                              void* d_out, int out_size, void* d_ws, size_t ws_size,
                              hipStream_t stream) {
    const float* x           = (const float*)d_in[0];
    const float* a_cache     = (const float*)d_in[1];
    const float* b_cache     = (const float*)d_in[2];
    const float* base_weight = (const float*)d_in[3];
    const float* scaling     = (const float*)d_in[4];
    const int*   q_start_loc = (const int*)d_in[5];
    const int*   q_seqlens   = (const int*)d_in[6];
    const int*   adapter_ids = (const int*)d_in[7];
    const int*   rank_offset = (const int*)d_in[8];
    const int*   ranks       = (const int*)d_in[9];

    const int num_seqs     = in_sizes[5];
    const int num_adapters = in_sizes[9];
    const int max_rank     = in_sizes[8] / num_adapters;        // 64
    const int cache_len    = num_adapters * max_rank;           // 512
    const int K            = in_sizes[1] / cache_len;           // 4096
    const int N            = in_sizes[2] / cache_len;           // 4096
    const int T            = in_sizes[0] / K;                   // 8192

    float* xa  = (float*)d_ws;                                   // T * max_rank floats
    float* out = (float*)d_out;

    dim3 g1(T / XA_BM);
    lora_xa_kernel<<<g1, 256, 0, stream>>>(x, a_cache, q_start_loc, q_seqlens,
                                           adapter_ids, rank_offset, ranks, scaling,
                                           xa, K, num_seqs, max_rank);

    dim3 g2(N / BN, T / BM);   // n-fastest: x row-panel reused from L2 across n-blocks
    lora_main_kernel<<<g2, 256, 0, stream>>>(x, base_weight, b_cache, xa,
                                             q_start_loc, q_seqlens, adapter_ids,
                                             rank_offset, out, K, N, num_seqs, max_rank);
}